// BCESurvLoss_75634374083220
// MI455X (gfx1250) — compile-verified
//
#include <hip/hip_runtime.h>

// BCE survival loss, MI455X (gfx1250, wave32).
// Streaming-bandwidth bound: 67 MB @ 23.3 TB/s ~= 2.9 us floor.
// bce = w_j * (mask * softplus(x) - y * x), 2 transcendentals/elt.
// Cross-lane reduction via V_WMMA_F32_16X16X4_F32 with all-ones B (exact f32).

#define T_BINS 128
#define BLOCK  256
#define WPB    (BLOCK / 32)   // waves per block
#define GRID_MAX 1024

typedef __attribute__((ext_vector_type(2))) float v2f;
typedef __attribute__((ext_vector_type(4))) float v4f;
typedef __attribute__((ext_vector_type(8))) float v8f;

// Exact 32-lane f32 sum using WMMA f32 16x16x4 with all-ones B.
// A[m][k] layout (ISA 7.12.2): lane L<16 holds A[L][0..1] in v0,v1;
// lane L>=16 holds A[L-16][2..3]. We place acc in v0, 0 in v1, so
// D[m][n] = acc_m + acc_{m+16} for EVERY n (all-ones B => layout independent).
// D components in lane L: VGPR r = D[r + 8*(L>=16)][L%16]; summing the 8
// components and one shfl_xor(16) yields the full 32-lane sum in all lanes.
__device__ __forceinline__ float wave_sum32(float v) {
    v2f a; a[0] = v;    a[1] = 0.0f;
    v2f b; b[0] = 1.0f; b[1] = 1.0f;
    v8f c = {};
    c = __builtin_amdgcn_wmma_f32_16x16x4_f32(
        /*neg_a=*/false, a, /*neg_b=*/false, b,
        /*c_mod=*/(short)0, c, /*reuse_a=*/false, /*reuse_b=*/false);
    float s = ((c[0] + c[1]) + (c[2] + c[3])) + ((c[4] + c[5]) + (c[6] + c[7]));
    s += __shfl_xor(s, 16, 32);
    return s;
}

__global__ __launch_bounds__(BLOCK) void survloss_partial_kernel(
    const float* __restrict__ preds,   // [N, 128]
    const float* __restrict__ weight,  // [128]
    const float* __restrict__ sw,      // [N]
    const int*   __restrict__ td,      // [N]
    const int*   __restrict__ te,      // [N]
    float*       __restrict__ partials,// [2 * gridDim.x] in d_ws
    int N)
{
    const int lane   = threadIdx.x & 31;
    const int wave   = threadIdx.x >> 5;
    const int waveId = blockIdx.x * WPB + wave;
    const int nWaves = gridDim.x * WPB;

    // Per-lane column weights, register-resident for the whole row loop.
    const v4f wv = *reinterpret_cast<const v4f*>(weight + 4 * lane);
    const int jBase = 4 * lane;

    float numAcc = 0.0f;   // sum over rows of (sw/maskSum) * sum_j bce_j
    float denAcc = 0.0f;   // 32x the sw sum (every lane adds sw); fixed by /32

    for (int r = waveId; r < N; r += nWaves) {
        int d = td[r];
        d = d < 0 ? 0 : (d > (T_BINS - 1) ? (T_BINS - 1) : d);
        const bool ev   = (te[r] != 0);
        const float swr = sw[r];
        const float scale = swr * (ev ? (1.0f / (float)T_BINS)
                                      : 1.0f / (float)(d + 1));

        // One coalesced 512B row per wave; non-temporal (single pass).
        const v4f x = __builtin_nontemporal_load(
            reinterpret_cast<const v4f*>(preds + (size_t)r * T_BINS + jBase));

        if (r + nWaves < N) {
            __builtin_prefetch(preds + (size_t)(r + nWaves) * T_BINS + jBase, 0, 1);
        }

        float rowPart = 0.0f;
#pragma unroll
        for (int c = 0; c < 4; ++c) {
            const int   j  = jBase + c;
            const float xv = x[c];
            const float le = (j <= d) ? 1.0f : 0.0f;
            const float y  = ev ? ((j < d) ? 1.0f : 0.0f) : le;
            const float mk = ev ? 1.0f : le;
            // softplus(x) = max(x,0) + log1p(exp(-|x|)); 2 transcendentals.
            const float L  = __logf(1.0f + __expf(-fabsf(xv)));
            const float sp = fmaxf(xv, 0.0f) + L;
            rowPart = fmaf(wv[c], mk * sp - y * xv, rowPart);
        }
        numAcc = fmaf(scale, rowPart, numAcc);
        denAcc += swr;
    }

    // Exact WMMA wave reductions (EXEC all-ones here: uniform control flow).
    const float wNum = wave_sum32(numAcc);
    const float wDen = wave_sum32(denAcc) * (1.0f / 32.0f);

    __shared__ float sNum[WPB];
    __shared__ float sDen[WPB];
    if (lane == 0) { sNum[wave] = wNum; sDen[wave] = wDen; }
    __syncthreads();
    if (threadIdx.x == 0) {
        float n = 0.0f, dd = 0.0f;
#pragma unroll
        for (int i = 0; i < WPB; ++i) { n += sNum[i]; dd += sDen[i]; }
        partials[blockIdx.x]             = n;
        partials[gridDim.x + blockIdx.x] = dd;
    }
}

__global__ __launch_bounds__(32) void survloss_final_kernel(
    const float* __restrict__ partials, float* __restrict__ out, int nparts)
{
    float n = 0.0f, d = 0.0f;
    for (int i = threadIdx.x; i < nparts; i += 32) {
        n += partials[i];
        d += partials[nparts + i];
    }
    n = wave_sum32(n);
    d = wave_sum32(d);
    if (threadIdx.x == 0) {
        out[0] = n / fmaxf(d, 1e-9f);
    }
}

extern "C" void kernel_launch(void* const* d_in, const int* in_sizes, int n_in,
                              void* d_out, int out_size, void* d_ws, size_t ws_size,
                              hipStream_t stream) {
    const float* preds  = (const float*)d_in[0];
    const float* weight = (const float*)d_in[1];
    const float* sw     = (const float*)d_in[2];
    const int*   td     = (const int*)d_in[3];
    const int*   te     = (const int*)d_in[4];
    const int N = in_sizes[2];  // sample_weight length

    int grid = GRID_MAX;
    const size_t need = (size_t)grid * 2 * sizeof(float);
    if (ws_size < need) {
        grid = (int)(ws_size / (2 * sizeof(float)));
        if (grid < 1) grid = 1;
    }

    float* partials = (float*)d_ws;
    survloss_partial_kernel<<<grid, BLOCK, 0, stream>>>(
        preds, weight, sw, td, te, partials, N);
    survloss_final_kernel<<<1, 32, 0, stream>>>(partials, (float*)d_out, grid);
}